// InterfaceBoundaryLoss_71416716197972
// MI455X (gfx1250) — compile-verified
//
#include <hip/hip_runtime.h>

// Problem constants from the reference
#define B_    4
#define H_    2048
#define W_    2048
#define INV_DX 100.0f   // 1/DX = 1/DY
#define E_IN_  2.0f
#define CONST_ 1.0f
#define WEIGHT_ 1.0f

typedef __attribute__((ext_vector_type(2))) float v2f;
typedef __attribute__((ext_vector_type(8))) float v8f;

__global__ void ibl_init(float* ws_sum, unsigned int* ws_cnt) {
    *ws_sum = 0.0f;
    *ws_cnt = 0u;
}

__global__ __launch_bounds__(256) void ibl_main(
    const float* __restrict__ sub1, const float* __restrict__ sub2,
    const unsigned char* __restrict__ intf, const unsigned char* __restrict__ bnd,
    float* __restrict__ ws_sum, unsigned int* __restrict__ ws_cnt)
{
    const int tid = blockIdx.x * blockDim.x + threadIdx.x;
    const int i  = tid >> 7;          // row: tid / (W/16)
    const int j0 = (tid & 127) << 4;  // first of 16 consecutive columns

    float acc = 0.0f;
    unsigned int cnt = 0u;

    // Coalesced 16-byte read of the boundary mask (dominant memory traffic).
    const uint4 bq = *(const uint4*)(bnd + (size_t)i * W_ + j0);
    if (bq.x | bq.y | bq.z | bq.w) {
        unsigned int bw[4] = { bq.x, bq.y, bq.z, bq.w };
        #pragma unroll 1
        for (int k = 0; k < 16; ++k) {
            if ((bw[k >> 2] >> ((k & 3) * 8)) & 0xFFu) {
                const int j  = j0 + k;
                const int im = (i > 0)      ? i - 1 : 0;       // edge-replicated shifts
                const int ip = (i < H_ - 1) ? i + 1 : H_ - 1;
                const int jm = (j > 0)      ? j - 1 : 0;
                const int jp = (j < W_ - 1) ? j + 1 : W_ - 1;

                const size_t pc = (size_t)i  * W_ + j;
                const size_t pu = (size_t)im * W_ + j;
                const size_t pd = (size_t)ip * W_ + j;
                const size_t pl = (size_t)i  * W_ + jm;
                const size_t pr = (size_t)i  * W_ + jp;

                // ~mask == ~mask_shift  <=>  mask == mask_shift : shared by sub1 & sub2
                const bool mc = intf[pc] != 0;
                const bool cu = ((intf[pu] != 0) == mc);
                const bool cd = ((intf[pd] != 0) == mc);
                const bool cl = ((intf[pl] != 0) == mc);
                const bool cr = ((intf[pr] != 0) == mc);
                const bool valid = (i >= 1) && (j >= 1);

                cnt++;  // nb counts boundary pixels once (not per batch)

                #pragma unroll
                for (int b = 0; b < B_; ++b) {
                    const size_t o = (size_t)b * ((size_t)H_ * W_);
                    const float s1 = sub1[o + pc];
                    const float s2 = sub2[o + pc];
                    const float d  = s1 - s2;
                    acc += d * d;

                    float gx1 = 0.0f, gy1 = 0.0f, gx2 = 0.0f, gy2 = 0.0f;
                    if (valid) {
                        if (cu) {
                            gx1 = (s1 - sub1[o + pu]) * INV_DX;
                            gx2 = (s2 - sub2[o + pu]) * INV_DX;
                        } else if (cd) {
                            gx1 = (s1 - sub1[o + pd]) * INV_DX;
                            gx2 = (s2 - sub2[o + pd]) * INV_DX;
                        }
                        if (cl) {
                            gy1 = (s1 - sub1[o + pl]) * INV_DX;
                            gy2 = (s2 - sub2[o + pl]) * INV_DX;
                        } else if (cr) {
                            gy1 = (s1 - sub1[o + pr]) * INV_DX;
                            gy2 = (s2 - sub2[o + pr]) * INV_DX;
                        }
                    }
                    float t;
                    t = E_IN_ * gx1 - CONST_; acc += t * t;
                    t = E_IN_ * gy1 - CONST_; acc += t * t;
                    t = E_IN_ * gx2 - CONST_; acc += t * t;
                    t = E_IN_ * gy2 - CONST_; acc += t * t;
                }
            }
        }
    }

    // ---- wave32 reduction of `acc` via the matrix pipe ----
    // A (16x4) holds each lane's {acc, 0}; B = all-ones (layout-invariant).
    // D[m,n] = row-sum of A. Summing a lane's 8 D-VGPRs yields the half-wave
    // total (lanes 0-15: rows 0-7, lanes 16-31: rows 8-15); one xor-16 shuffle
    // combines the halves. EXEC is all-ones here (uniform control flow).
    v2f a; a[0] = acc;  a[1] = 0.0f;
    v2f bo; bo[0] = 1.0f; bo[1] = 1.0f;
    v8f cz = {};
    v8f dm = __builtin_amdgcn_wmma_f32_16x16x4_f32(
        /*neg_a=*/false, a, /*neg_b=*/false, bo,
        /*c_mod=*/(short)0, cz, /*reuse_a=*/false, /*reuse_b=*/false);
    float s8 = dm[0] + dm[1] + dm[2] + dm[3] + dm[4] + dm[5] + dm[6] + dm[7];
    float wsum = s8 + __shfl_xor(s8, 16, 32);

    int wcnt = (int)cnt;
    #pragma unroll
    for (int off = 16; off > 0; off >>= 1)
        wcnt += __shfl_xor(wcnt, off, 32);

    // ---- block reduction through LDS, one atomic pair per block ----
    __shared__ float        ls[8];
    __shared__ unsigned int lc[8];
    const int lane = threadIdx.x & 31;
    const int wv   = threadIdx.x >> 5;
    if (lane == 0) { ls[wv] = wsum; lc[wv] = (unsigned int)wcnt; }
    __syncthreads();
    if (threadIdx.x == 0) {
        float s = 0.0f; unsigned int c = 0u;
        #pragma unroll
        for (int w = 0; w < 8; ++w) { s += ls[w]; c += lc[w]; }
        atomicAdd(ws_sum, s);
        atomicAdd(ws_cnt, c);
    }
}

__global__ void ibl_fin(const float* ws_sum, const unsigned int* ws_cnt,
                        float* out) {
    const float nb = (float)(*ws_cnt);
    out[0] = (*ws_sum / ((float)B_ * nb)) * WEIGHT_;
}

extern "C" void kernel_launch(void* const* d_in, const int* in_sizes, int n_in,
                              void* d_out, int out_size, void* d_ws, size_t ws_size,
                              hipStream_t stream) {
    (void)in_sizes; (void)n_in; (void)out_size; (void)ws_size;
    const float*         sub1 = (const float*)d_in[0];
    const float*         sub2 = (const float*)d_in[1];
    const unsigned char* intf = (const unsigned char*)d_in[2];  // jnp bool -> 1 byte
    const unsigned char* bnd  = (const unsigned char*)d_in[3];

    float*        ws_sum = (float*)d_ws;
    unsigned int* ws_cnt = (unsigned int*)((char*)d_ws + sizeof(float));

    ibl_init<<<1, 1, 0, stream>>>(ws_sum, ws_cnt);

    const int n_threads = (H_ * W_) / 16;          // 262144: one uint4 / thread
    ibl_main<<<n_threads / 256, 256, 0, stream>>>(sub1, sub2, intf, bnd,
                                                  ws_sum, ws_cnt);

    ibl_fin<<<1, 1, 0, stream>>>(ws_sum, ws_cnt, (float*)d_out);
}